// GCN_6244882448867
// MI455X (gfx1250) — compile-verified
//
#include <hip/hip_runtime.h>
#include <math.h>

typedef float v2f __attribute__((ext_vector_type(2)));
typedef float v8f __attribute__((ext_vector_type(8)));

#define NEG_SLOPE 0.2f
#define BN_EPS 1e-5f

static inline int ceil_div_l(long a, long b) { return (int)((a + b - 1) / b); }

// ---------------------------------------------------------------------------
// WMMA f32 GEMM: C[M,O] = A[M,K] @ W[K,O] (+ bias). One wave = one 16x16 tile.
// A 16x4 layout: lanes0-15 hold M=lane {K=k,k+1}; lanes16-31 hold {K=k+2,k+3}.
// B 4x16 layout: lanes hold column N=lane&15; half selects K pair.
// C/D: VGPR i holds row M = i + 8*(lane>>4), col N = lane&15.
// ---------------------------------------------------------------------------
__global__ __launch_bounds__(256) void k_wmma_gemm(
    const float* __restrict__ A, const float* __restrict__ W,
    const float* __restrict__ bias, float* __restrict__ C,
    int M, int K, int O) {
  const int wpb = blockDim.x >> 5;
  int wave = blockIdx.x * wpb + (threadIdx.x >> 5);
  int tiles_n = O >> 4;
  int tiles_m = M >> 4;
  if (wave >= tiles_m * tiles_n) return;       // wave-uniform: EXEC stays all-1s
  int tm = wave / tiles_n, tn = wave % tiles_n;
  int lane = threadIdx.x & 31;
  int half = lane >> 4;
  int l = lane & 15;
  int mrow = (tm << 4) + l;
  int ncol = (tn << 4) + l;
  const float* arow = A + (size_t)mrow * K;
  v8f acc = {};
  for (int kk = 0; kk < K; kk += 4) {
    int kb = kk + (half << 1);
    v2f a = *reinterpret_cast<const v2f*>(arow + kb);      // contiguous, 8B aligned
    v2f b;
    b.x = W[(size_t)kb * O + ncol];
    b.y = W[(size_t)(kb + 1) * O + ncol];
    acc = __builtin_amdgcn_wmma_f32_16x16x4_f32(
        /*neg_a=*/false, a, /*neg_b=*/false, b,
        /*c_mod=*/(short)0, acc, /*reuse_a=*/false, /*reuse_b=*/false);
  }
  float bb = bias ? bias[ncol] : 0.0f;
#pragma unroll
  for (int i = 0; i < 8; ++i) {
    int row = (tm << 4) + i + (half << 3);
    C[(size_t)row * O + ncol] = acc[i] + bb;
  }
}

// ---------------------------------------------------------------------------
// Degree / symmetric GCN norm
// ---------------------------------------------------------------------------
__global__ void k_degree(const int* __restrict__ dst, float* __restrict__ deg, int E) {
  int i = blockIdx.x * blockDim.x + threadIdx.x;
  if (i < E) atomicAdd(&deg[dst[i]], 1.0f);
}
__global__ void k_dinv(float* __restrict__ deg, int n) {
  int i = blockIdx.x * blockDim.x + threadIdx.x;
  if (i < n) deg[i] = rsqrtf(deg[i] + 1.0f);
}

// GCN self-loop term: out[n,f] = hw[n,f]*dinv[n]^2 + b[f]
__global__ void k_gcn_self(const float* __restrict__ hw, const float* __restrict__ dinv,
                           const float* __restrict__ bias, float* __restrict__ out,
                           int n, int O) {
  size_t i = (size_t)blockIdx.x * blockDim.x + threadIdx.x;
  if (i >= (size_t)n * O) return;
  int row = (int)(i / O), c = (int)(i % O);
  float sn = dinv[row];
  out[i] = hw[i] * sn * sn + bias[c];
}

// GCN neighbor aggregation: out[dst] += hw[src] * dinv[src]*dinv[dst].
// Wave per edge, lane per feature: coalesced gather, f32 global atomics scatter.
__global__ void k_gcn_agg(const float* __restrict__ hw, const int* __restrict__ src,
                          const int* __restrict__ dst, const float* __restrict__ dinv,
                          float* __restrict__ out, int E, int O) {
  int wpb = blockDim.x >> 5;
  int nw = gridDim.x * wpb;
  int lane = threadIdx.x & 31;
  for (int e = blockIdx.x * wpb + (threadIdx.x >> 5); e < E; e += nw) {
    int si = src[e], di = dst[e];
    float norm = dinv[si] * dinv[di];
    const float* hs = hw + (size_t)si * O;
    float* od = out + (size_t)di * O;
    for (int f = lane; f < O; f += 32) atomicAdd(&od[f], hs[f] * norm);
  }
}

// ---------------------------------------------------------------------------
// GAT: per-row attention scores (wave per row, shuffle reduce)
// ---------------------------------------------------------------------------
__global__ void k_gat_scores(const float* __restrict__ hw, const float* __restrict__ as,
                             const float* __restrict__ ad, float* __restrict__ s,
                             float* __restrict__ d, int n, int O) {
  int wpb = blockDim.x >> 5;
  int row = blockIdx.x * wpb + (threadIdx.x >> 5);
  if (row >= n) return;
  int lane = threadIdx.x & 31;
  float accs = 0.f, accd = 0.f;
  for (int f = lane; f < O; f += 32) {
    float v = hw[(size_t)row * O + f];
    accs += v * as[f];
    accd += v * ad[f];
  }
  for (int m = 16; m; m >>= 1) {
    accs += __shfl_xor(accs, m, 32);
    accd += __shfl_xor(accd, m, 32);
  }
  if (lane == 0) { s[row] = accs; d[row] = accd; }
}

// order-preserving float<->uint encoding for atomic max over signed floats
__device__ __forceinline__ unsigned f2o(float f) {
  unsigned b = __float_as_uint(f);
  return b ^ ((b >> 31) ? 0xFFFFFFFFu : 0x80000000u);
}
__device__ __forceinline__ float o2f(unsigned u) {
  return __uint_as_float((u & 0x80000000u) ? (u ^ 0x80000000u) : ~u);
}

// pass 1: segment max of leaky_relu(s[src]+d[dst]) over combined (edges + self-loops)
__global__ void k_gat_max(const int* __restrict__ src, const int* __restrict__ dst,
                          const float* __restrict__ s, const float* __restrict__ d,
                          unsigned* __restrict__ m, int E, int n) {
  int i = blockIdx.x * blockDim.x + threadIdx.x;
  if (i >= E + n) return;
  int si = (i < E) ? src[i] : (i - E);
  int di = (i < E) ? dst[i] : (i - E);
  float e = s[si] + d[di];
  e = (e >= 0.f) ? e : NEG_SLOPE * e;
  atomicMax(&m[di], f2o(e));
}

// pass 2: den[dst]+=exp(e-m); out[dst,:]+=exp(e-m)*hw[src,:]
__global__ void k_gat_acc(const float* __restrict__ hw, const int* __restrict__ src,
                          const int* __restrict__ dst, const float* __restrict__ s,
                          const float* __restrict__ d, const unsigned* __restrict__ m,
                          float* __restrict__ den, float* __restrict__ out,
                          int E, int n, int O) {
  int wpb = blockDim.x >> 5;
  int nw = gridDim.x * wpb;
  int lane = threadIdx.x & 31;
  int tot = E + n;
  for (int i = blockIdx.x * wpb + (threadIdx.x >> 5); i < tot; i += nw) {
    int si = (i < E) ? src[i] : (i - E);
    int di = (i < E) ? dst[i] : (i - E);
    float e = s[si] + d[di];
    e = (e >= 0.f) ? e : NEG_SLOPE * e;
    float wexp = expf(e - o2f(m[di]));
    if (lane == 0) atomicAdd(&den[di], wexp);
    const float* hs = hw + (size_t)si * O;
    float* od = out + (size_t)di * O;
    for (int f = lane; f < O; f += 32) atomicAdd(&od[f], wexp * hs[f]);
  }
}

// pass 3: normalize + bias
__global__ void k_gat_fin(float* __restrict__ out, const float* __restrict__ den,
                          const float* __restrict__ bias, int n, int O) {
  size_t i = (size_t)blockIdx.x * blockDim.x + threadIdx.x;
  if (i >= (size_t)n * O) return;
  int row = (int)(i / O), c = (int)(i % O);
  out[i] = out[i] / den[row] + bias[c];
}

// ---------------------------------------------------------------------------
// BatchNorm (training-mode batch stats over rows). LDS f32 atomics for partials.
// ---------------------------------------------------------------------------
__global__ __launch_bounds__(256) void k_bn_stats(const float* __restrict__ h,
                                                  float* __restrict__ gsum,
                                                  float* __restrict__ gsq, int n, int O) {
  __shared__ float ls[256], lq[256];
  for (int c = threadIdx.x; c < O; c += blockDim.x) { ls[c] = 0.f; lq[c] = 0.f; }
  __syncthreads();
  size_t tot = (size_t)n * O;
  size_t stride = (size_t)gridDim.x * blockDim.x;
  for (size_t i = (size_t)blockIdx.x * blockDim.x + threadIdx.x; i < tot; i += stride) {
    float v = h[i];
    int c = (int)(i % O);
    atomicAdd(&ls[c], v);        // ds_add_f32
    atomicAdd(&lq[c], v * v);
  }
  __syncthreads();
  for (int c = threadIdx.x; c < O; c += blockDim.x) {
    atomicAdd(&gsum[c], ls[c]);
    atomicAdd(&gsq[c], lq[c]);
  }
}
__global__ void k_bn_fin(float* __restrict__ gsum, float* __restrict__ gsq, int n, int O) {
  int c = blockIdx.x * blockDim.x + threadIdx.x;
  if (c >= O) return;
  float m = gsum[c] / (float)n;
  float v = gsq[c] / (float)n - m * m;
  gsum[c] = m;
  gsq[c] = rsqrtf(v + BN_EPS);
}
// act: 0=none 1=tanh 2=relu; dst_stride lets branch outputs land in xi slices
__global__ void k_bn_apply(const float* __restrict__ src, float* __restrict__ dstp,
                           const float* __restrict__ mean, const float* __restrict__ rstd,
                           const float* __restrict__ g, const float* __restrict__ be,
                           int n, int O, int dst_stride, int act) {
  size_t i = (size_t)blockIdx.x * blockDim.x + threadIdx.x;
  if (i >= (size_t)n * O) return;
  int row = (int)(i / O), c = (int)(i % O);
  float v = (src[i] - mean[c]) * rstd[c] * g[c] + be[c];
  if (act == 1) v = tanhf(v);
  else if (act == 2) v = fmaxf(v, 0.f);
  dstp[(size_t)row * dst_stride + c] = v;
}

// ---------------------------------------------------------------------------
// global mean pool and output head
// ---------------------------------------------------------------------------
__global__ void k_pool(const float* __restrict__ xa, const int* __restrict__ batch,
                       float* __restrict__ pool, float* __restrict__ cnt, int n, int O) {
  int wpb = blockDim.x >> 5;
  int row = blockIdx.x * wpb + (threadIdx.x >> 5);
  if (row >= n) return;
  int lane = threadIdx.x & 31;
  int g = batch[row];
  if (lane == 0) atomicAdd(&cnt[g], 1.0f);
  for (int f = lane; f < O; f += 32)
    atomicAdd(&pool[(size_t)g * O + f], xa[(size_t)row * O + f]);
}
__global__ void k_pool_div(const float* __restrict__ pool, const float* __restrict__ cnt,
                           float* __restrict__ out, int G, int O) {
  int i = blockIdx.x * blockDim.x + threadIdx.x;
  if (i >= G * O) return;
  out[i] = pool[i] / fmaxf(cnt[i / O], 1.0f);
}
__global__ void k_out(const float* __restrict__ h, const float* __restrict__ w,
                      const float* __restrict__ b, float* __restrict__ out, int G) {
  int wpb = blockDim.x >> 5;
  int g = blockIdx.x * wpb + (threadIdx.x >> 5);
  if (g >= G) return;
  int lane = threadIdx.x & 31;
  float acc = h[(size_t)g * 32 + lane] * w[lane];   // K == 32 == wave32
  for (int m = 16; m; m >>= 1) acc += __shfl_xor(acc, m, 32);
  if (lane == 0) out[g] = acc + b[0];
}

// ---------------------------------------------------------------------------
// host-side layer helpers (all launches on `stream`, graph-capture safe)
// ---------------------------------------------------------------------------
static void gemm(const float* A, const float* Wm, const float* bias, float* C,
                 int M, int K, int O, hipStream_t s) {
  int tiles = (M / 16) * (O / 16);
  k_wmma_gemm<<<ceil_div_l(tiles, 8), 256, 0, s>>>(A, Wm, bias, C, M, K, O);
}
static void bn(const float* src, float* dstp, const float* g, const float* be,
               float* sum, float* sq, int n, int O, int dst_stride, int act,
               hipStream_t s) {
  hipMemsetAsync(sum, 0, O * sizeof(float), s);
  hipMemsetAsync(sq, 0, O * sizeof(float), s);
  long blocks = ((long)n * O + 255) / 256;
  if (blocks > 1024) blocks = 1024;
  k_bn_stats<<<(int)blocks, 256, 0, s>>>(src, sum, sq, n, O);
  k_bn_fin<<<ceil_div_l(O, 64), 64, 0, s>>>(sum, sq, n, O);
  k_bn_apply<<<ceil_div_l((long)n * O, 256), 256, 0, s>>>(src, dstp, sum, sq, g, be,
                                                          n, O, dst_stride, act);
}
static void gcn(const float* h, int K, int O, const float* Wm, const float* bias,
                const float* dinv, const int* src, const int* dst, int N, int E,
                float* hw, float* out, hipStream_t s) {
  gemm(h, Wm, nullptr, hw, N, K, O, s);
  k_gcn_self<<<ceil_div_l((long)N * O, 256), 256, 0, s>>>(hw, dinv, bias, out, N, O);
  k_gcn_agg<<<ceil_div_l(E, 8), 256, 0, s>>>(hw, src, dst, dinv, out, E, O);
}
static void gat(const float* h, int K, int O, const float* Wm, const float* as,
                const float* ad, const float* bias, const int* src, const int* dst,
                int N, int E, float* hw, float* sv, float* dv, unsigned* mb,
                float* den, float* out, hipStream_t s) {
  gemm(h, Wm, nullptr, hw, N, K, O, s);
  k_gat_scores<<<ceil_div_l(N, 8), 256, 0, s>>>(hw, as, ad, sv, dv, N, O);
  hipMemsetAsync(mb, 0, N * sizeof(unsigned), s);
  int tot = E + N;
  k_gat_max<<<ceil_div_l(tot, 256), 256, 0, s>>>(src, dst, sv, dv, mb, E, N);
  hipMemsetAsync(den, 0, N * sizeof(float), s);
  hipMemsetAsync(out, 0, (size_t)N * O * sizeof(float), s);
  k_gat_acc<<<ceil_div_l(tot, 8), 256, 0, s>>>(hw, src, dst, sv, dv, mb, den, out, E, N, O);
  k_gat_fin<<<ceil_div_l((long)N * O, 256), 256, 0, s>>>(out, den, bias, N, O);
}

extern "C" void kernel_launch(void* const* d_in, const int* in_sizes, int n_in,
                              void* d_out, int out_size, void* d_ws, size_t ws_size,
                              hipStream_t stream) {
  (void)n_in; (void)ws_size;
  const float* x = (const float*)d_in[0];
  const int* ei = (const int*)d_in[1];
  const int* batch = (const int*)d_in[2];
  const int N = in_sizes[2];
  const int E = in_sizes[1] / 2;
  const int D = in_sizes[0] / N;   // 128
  const int G = out_size;          // 256
  const int* src = ei;
  const int* dst = ei + E;
  auto p = [&](int i) { return (const float*)d_in[i]; };

  // workspace carve
  float* w = (float*)d_ws;
  auto take = [&](size_t ne) { float* r = w; w += ne; return r; };
  float* t1   = take((size_t)N * 128);   // pre-agg hw
  float* t2   = take((size_t)N * 128);   // layer output (pre-BN)
  float* xl   = take((size_t)N * 128);
  float* xm   = take((size_t)N * 128);
  float* xr   = take((size_t)N * 128);
  float* xi   = take((size_t)N * 224);   // concat target
  float* xa   = take((size_t)N * 128);
  float* dinv = take(N);
  float* sv   = take(N);
  float* dvv  = take(N);
  unsigned* mb = (unsigned*)take(N);
  float* den  = take(N);
  float* cnt  = take(G);
  float* pool = take((size_t)G * 128);
  float* fa   = take((size_t)G * 256);
  float* fb   = take((size_t)G * 256);
  float* ssum = take(256);
  float* ssq  = take(256);

  // GCN normalization: dinv = rsqrt(in_degree + 1)
  hipMemsetAsync(dinv, 0, N * sizeof(float), stream);
  k_degree<<<ceil_div_l(E, 256), 256, 0, stream>>>(dst, dinv, E);
  k_dinv<<<ceil_div_l(N, 256), 256, 0, stream>>>(dinv, N);

  // branch L: gcn(128) -> bn tanh -> gat(128) -> bn tanh -> xi[:,0:128]
  gcn(x, D, 128, p(3), p(4), dinv, src, dst, N, E, t1, t2, stream);
  bn(t2, xl, p(5), p(6), ssum, ssq, N, 128, 128, 1, stream);
  gat(xl, 128, 128, p(7), p(8), p(9), p(10), src, dst, N, E, t1, sv, dvv, mb, den, t2, stream);
  bn(t2, xi + 0, p(11), p(12), ssum, ssq, N, 128, 224, 1, stream);

  // branch M: gcn(128) -> gcn(64) -> gat(64) -> xi[:,128:192]
  gcn(x, D, 128, p(13), p(14), dinv, src, dst, N, E, t1, t2, stream);
  bn(t2, xm, p(15), p(16), ssum, ssq, N, 128, 128, 1, stream);
  gcn(xm, 128, 64, p(17), p(18), dinv, src, dst, N, E, t1, t2, stream);
  bn(t2, xm, p(19), p(20), ssum, ssq, N, 64, 64, 1, stream);
  gat(xm, 64, 64, p(21), p(22), p(23), p(24), src, dst, N, E, t1, sv, dvv, mb, den, t2, stream);
  bn(t2, xi + 128, p(25), p(26), ssum, ssq, N, 64, 224, 1, stream);

  // branch R: gcn(128) -> gcn(64) -> gcn(32) -> gat(32) -> xi[:,192:224]
  gcn(x, D, 128, p(27), p(28), dinv, src, dst, N, E, t1, t2, stream);
  bn(t2, xr, p(29), p(30), ssum, ssq, N, 128, 128, 2, stream);
  gcn(xr, 128, 64, p(31), p(32), dinv, src, dst, N, E, t1, t2, stream);
  bn(t2, xr, p(33), p(34), ssum, ssq, N, 64, 64, 2, stream);
  gcn(xr, 64, 32, p(35), p(36), dinv, src, dst, N, E, t1, t2, stream);
  bn(t2, xr, p(37), p(38), ssum, ssq, N, 32, 32, 2, stream);
  gat(xr, 32, 32, p(39), p(40), p(41), p(42), src, dst, N, E, t1, sv, dvv, mb, den, t2, stream);
  bn(t2, xi + 192, p(43), p(44), ssum, ssq, N, 32, 224, 2, stream);

  // fused head: gat_all(224 -> 128) -> bn relu -> xa
  gat(xi, 224, 128, p(45), p(46), p(47), p(48), src, dst, N, E, t1, sv, dvv, mb, den, t2, stream);
  bn(t2, xa, p(49), p(50), ssum, ssq, N, 128, 128, 2, stream);

  // global mean pool -> fa[G,128]
  hipMemsetAsync(cnt, 0, G * sizeof(float), stream);
  hipMemsetAsync(pool, 0, (size_t)G * 128 * sizeof(float), stream);
  k_pool<<<ceil_div_l(N, 8), 256, 0, stream>>>(xa, batch, pool, cnt, N, 128);
  k_pool_div<<<ceil_div_l((long)G * 128, 256), 256, 0, stream>>>(pool, cnt, fa, G, 128);

  // MLP head: gemm+bias -> bn relu, 4x
  gemm(fa, p(51), p(52), fb, G, 128, 256, stream);
  bn(fb, fa, p(53), p(54), ssum, ssq, G, 256, 256, 2, stream);
  gemm(fa, p(55), p(56), fb, G, 256, 128, stream);
  bn(fb, fa, p(57), p(58), ssum, ssq, G, 128, 128, 2, stream);
  gemm(fa, p(59), p(60), fb, G, 128, 64, stream);
  bn(fb, fa, p(61), p(62), ssum, ssq, G, 64, 64, 2, stream);
  gemm(fa, p(63), p(64), fb, G, 64, 32, stream);
  bn(fb, fa, p(65), p(66), ssum, ssq, G, 32, 32, 2, stream);

  // output head [G,32] @ [32,1] + b
  k_out<<<ceil_div_l(G, 8), 256, 0, stream>>>(fa, p(67), p(68), (float*)d_out, G);
}